// Fair_Identity_Normalizer_Single_67791763800436
// MI455X (gfx1250) — compile-verified
//
#include <hip/hip_runtime.h>
#include <hip/hip_bf16.h>
#include <stdint.h>

// out[r, :] = (x[r, :] - mus[attr[r]]) / (log1p(exp(sigmas[attr[r]])) + EPS)
// MOMENTUM == 0 -> pure normalize. BW-bound: 512 MB / 23.3 TB/s ~= 22 us floor.
//
// Stage 1: tiny kernel folds the A=4 group constants (mu, 1/softplus) into d_ws
//          so the streaming kernel never touches transcendentals.
// Stage 2: 16 rows per block, double-buffered CDNA5 async pipeline:
//          global --(global_load_async_to_lds_b128, ASYNCcnt)--> LDS (2x4KB)
//          LDS    --(ds_load_b128)--> VGPRs --(sub/mul)--> NT global_store_b128
//          s_wait_asynccnt 1 overlaps the next row's DMA with this row's math.

#define FNORM_EPS 1e-6f
#define ROWS_PER_BLOCK 16

typedef float v4f __attribute__((ext_vector_type(4)));   // native vector for NT builtin

// ---- Stage 1: per-group constants -> d_ws: tbl[a]=mu, tbl[A+a]=1/(softplus+eps)
__global__ void fair_norm_stats_kernel(const float* __restrict__ mus,
                                       const float* __restrict__ sigmas,
                                       float* __restrict__ tbl, int A) {
    const int a = threadIdx.x;
    if (a < A) {
        tbl[a]     = mus[a];
        tbl[A + a] = 1.0f / (log1pf(expf(sigmas[a])) + FNORM_EPS);
    }
}

// ---- Stage 2: streaming normalize with double-buffered async global->LDS ----
__global__ __launch_bounds__(256) void fair_norm_main_kernel(
    const float* __restrict__ x,
    const int*   __restrict__ attr,   // int32 (JAX w/o x64 downgrades int64)
    const float* __restrict__ tbl,    // [2*A] from stage 1
    float*       __restrict__ out,
    int D, int A, int B)
{
    __shared__ __align__(16) float tile[2][1024];        // 2 x 4 KB double buffer

    const unsigned tid  = threadIdx.x;                   // 0..D/4-1 (256)
    const unsigned row0 = blockIdx.x * ROWS_PER_BLOCK;
    const int nrows = ((int)(B - row0) < ROWS_PER_BLOCK) ? (int)(B - row0)
                                                         : ROWS_PER_BLOCK;

    const unsigned lane_byte = tid * 16u;
    const unsigned lds0 = (unsigned)(uintptr_t)(&tile[0][0]) + lane_byte;
    const unsigned lds1 = (unsigned)(uintptr_t)(&tile[1][0]) + lane_byte;
    const unsigned rowbytes = (unsigned)D * 4u;          // 4 KB

    // prime the pipeline: row0 -> buf0
    {
        const unsigned g0 = row0 * rowbytes + lane_byte; // < 2^31 total bytes
        asm volatile("global_load_async_to_lds_b128 %0, %1, %2"
                     :: "v"(lds0), "v"(g0), "s"(x) : "memory");
    }

    for (int k = 0; k < nrows; ++k) {
        const unsigned row = row0 + (unsigned)k;

        if (k + 1 < nrows) {
            // prefetch row k+1 into the other buffer, keep 1 DMA in flight
            const unsigned gn   = (row + 1u) * rowbytes + lane_byte;
            const unsigned ldsn = ((k + 1) & 1) ? lds1 : lds0;
            asm volatile("global_load_async_to_lds_b128 %0, %1, %2"
                         :: "v"(ldsn), "v"(gn), "s"(x) : "memory");
            // in-order completion: cnt<=1 guarantees row k's buffer is resident
            asm volatile("s_wait_asynccnt 0x1" ::: "memory");
        } else {
            asm volatile("s_wait_asynccnt 0x0" ::: "memory");
        }

        // uniform per-row stats: 3 scalar loads, no math
        const int   a    = attr[row];
        const float mu   = tbl[a];
        const float rinv = tbl[A + a];

        const v4f v = ((const v4f*)tile[k & 1])[tid];    // ds_load_b128
        v4f r;
        r.x = (v.x - mu) * rinv;
        r.y = (v.y - mu) * rinv;
        r.z = (v.z - mu) * rinv;
        r.w = (v.w - mu) * rinv;

        v4f* dst = (v4f*)(out + (size_t)row * (size_t)D) + tid;
        __builtin_nontemporal_store(r, dst);             // th:TH_STORE_NT
    }
}

extern "C" void kernel_launch(void* const* d_in, const int* in_sizes, int n_in,
                              void* d_out, int out_size, void* d_ws, size_t ws_size,
                              hipStream_t stream) {
    const float* x      = (const float*)d_in[0];
    const int*   attr   = (const int*)  d_in[1];
    const float* mus    = (const float*)d_in[2];
    const float* sigmas = (const float*)d_in[3];
    float*       out    = (float*)      d_out;
    float*       tbl    = (float*)      d_ws;

    const int B = in_sizes[1];                 // 65536 rows
    const int D = in_sizes[0] / B;             // 1024 cols
    const int A = in_sizes[2];                 // 4 groups
    const int threads = D / 4;                 // 256: one 16B chunk per lane

    fair_norm_stats_kernel<<<1, 32, 0, stream>>>(mus, sigmas, tbl, A);

    const int grid = (B + ROWS_PER_BLOCK - 1) / ROWS_PER_BLOCK;   // 4096
    fair_norm_main_kernel<<<grid, threads, 0, stream>>>(x, attr, tbl, out, D, A, B);
}